// MsaHmmCell_51651276702563
// MI455X (gfx1250) — compile-verified
//
#include <hip/hip_runtime.h>

// ---------------- problem constants (from reference) ----------------
#define MDL   2
#define BATCH 64
#define TT    512
#define DD    26
#define DP    32          // D padded for K=32 emission GEMM
#define QQ    515
#define QP    528         // q padded to 33 tiles of 16
#define NTILES (QP / 16)  // 33
#define KBA   (QP / 4)    // 132 K-panels for transition GEMM
#define KBB   (DP / 4)    // 8 K-panels for emission GEMM
#define EPSF  1e-32f

typedef __attribute__((ext_vector_type(2))) float v2f;
typedef __attribute__((ext_vector_type(8))) float v8f;

// Packed WMMA B-operand layout: element (K-row rr in panel kb, N-col c) of
// tile nt lives at lane = (rr>=2)*16 + (c&15), slot = rr&1. One wave32 loads
// a whole 4x16 panel as 32 contiguous v2f (256 B, single global_load_b64).
__device__ __forceinline__ size_t apack_idx(int m, int r, int c) {
  const int nt = c >> 4, coln = c & 15;
  const int kb = r >> 2, rr = r & 3;
  const int lane = ((rr >> 1) << 4) + coln;
  return ((((size_t)m * NTILES + nt) * KBA + kb) * 32 + lane) * 2 + (rr & 1);
}
__device__ __forceinline__ size_t bpack_idx(int m, int q, int d) {
  const int nt = q >> 4, coln = q & 15;
  const int kb = d >> 2, rr = d & 3;
  const int lane = ((rr >> 1) << 4) + coln;
  return ((((size_t)m * NTILES + nt) * KBB + kb) * 32 + lane) * 2 + (rr & 1);
}

// ---------------- softmax of transition matrix A -> packed Apk -------------
__global__ __launch_bounds__(256) void softmaxA_kernel(
    const float* __restrict__ Alog, float* __restrict__ Apk) {
  const int row = blockIdx.x;               // 0 .. M*QP-1
  const int m = row / QP, r = row % QP;
  const int tid = threadIdx.x;
  __shared__ float red[256];
  if (r >= QQ) {                            // zero pad K-rows of every tile
    for (int c = tid; c < QP; c += 256) Apk[apack_idx(m, r, c)] = 0.f;
    return;
  }
  const float* in = Alog + ((size_t)m * QQ + r) * QQ;
  float mx = -1e30f;
  for (int c = tid; c < QQ; c += 256) mx = fmaxf(mx, in[c]);
  red[tid] = mx; __syncthreads();
  for (int s = 128; s > 0; s >>= 1) {
    if (tid < s) red[tid] = fmaxf(red[tid], red[tid + s]);
    __syncthreads();
  }
  mx = red[0]; __syncthreads();
  float sum = 0.f;
  for (int c = tid; c < QQ; c += 256) sum += __expf(in[c] - mx);
  red[tid] = sum; __syncthreads();
  for (int s = 128; s > 0; s >>= 1) {
    if (tid < s) red[tid] += red[tid + s];
    __syncthreads();
  }
  const float inv = 1.f / red[0];
  for (int c = tid; c < QP; c += 256)
    Apk[apack_idx(m, r, c)] = (c < QQ) ? __expf(in[c] - mx) * inv : 0.f;
}

// ---------------- softmax of emission matrix -> packed Bpk -----------------
__global__ __launch_bounds__(32) void softmaxB_kernel(
    const float* __restrict__ Blog, float* __restrict__ Bpk) {
  const int row = blockIdx.x;               // q-state index (N of B-operand)
  const int m = row / QP, r = row % QP;
  const int lane = threadIdx.x;             // lane == d (K of B-operand)
  if (r >= QQ) { Bpk[bpack_idx(m, r, lane)] = 0.f; return; }
  const float* in = Blog + ((size_t)m * QQ + r) * DD;
  float v = (lane < DD) ? in[lane] : -1e30f;
  float mx = v;
  for (int off = 16; off > 0; off >>= 1) mx = fmaxf(mx, __shfl_xor(mx, off));
  float e = (lane < DD) ? __expf(v - mx) : 0.f;
  float s = e;
  for (int off = 16; off > 0; off >>= 1) s += __shfl_xor(s, off);
  Bpk[bpack_idx(m, r, lane)] = e / s;       // pad d-rows store exact 0
}

// ---------------- softmax of initial distribution -> pip [M,QP] ------------
__global__ __launch_bounds__(256) void softmaxPi_kernel(
    const float* __restrict__ Ilog, float* __restrict__ pip) {
  const int m = blockIdx.x;
  const int tid = threadIdx.x;
  __shared__ float red[256];
  const float* in = Ilog + (size_t)m * QQ;
  float* outp = pip + (size_t)m * QP;
  float mx = -1e30f;
  for (int c = tid; c < QQ; c += 256) mx = fmaxf(mx, in[c]);
  red[tid] = mx; __syncthreads();
  for (int s = 128; s > 0; s >>= 1) {
    if (tid < s) red[tid] = fmaxf(red[tid], red[tid + s]);
    __syncthreads();
  }
  mx = red[0]; __syncthreads();
  float sum = 0.f;
  for (int c = tid; c < QQ; c += 256) sum += __expf(in[c] - mx);
  red[tid] = sum; __syncthreads();
  for (int s = 128; s > 0; s >>= 1) {
    if (tid < s) red[tid] += red[tid + s];
    __syncthreads();
  }
  const float inv = 1.f / red[0];
  for (int c = tid; c < QP; c += 256)
    outp[c] = (c < QQ) ? __expf(in[c] - mx) * inv : 0.f;
}

// ---------------- main forward recursion -----------------------------------
__global__ __launch_bounds__(256) void hmm_forward_kernel(
    const float* __restrict__ obs,   // [M,B,T,D]
    const float* __restrict__ Apk,   // packed transition B-operands
    const float* __restrict__ Bpk,   // packed emission B-operands
    const float* __restrict__ pip,   // [M,QP] padded initial dist
    float* __restrict__ out) {       // [T,M,B,QQ]
  const int m   = blockIdx.x >> 2;
  const int b0  = (blockIdx.x & 3) * 16;
  const int tid = threadIdx.x;
  const int lane = tid & 31;
  const int wave = tid >> 5;
  const int r16  = lane & 15;
  const int koff = (lane < 16) ? 0 : 2;   // fp32 A-operand K split per half-wave

  __shared__ float s_alpha[16 * QP];      // scaled forward variables (A-operand)
  __shared__ float s_obs[16 * DP];        // current obs tile, D zero-padded
  __shared__ float s_part[16 * 16];       // row-sum partials
  __shared__ float s_S[16];               // per-row scale factor
  __shared__ float s_ll[16];              // running log-likelihood per row

  if (tid < 16) s_ll[tid] = 0.f;

  const float* piM = pip + (size_t)m * QP;

  for (int t = 0; t < TT; ++t) {
    // stage obs tile (16 rows x DP) into LDS, zero-padding d >= DD
    for (int i = tid; i < 16 * DP; i += 256) {
      const int r = i >> 5, d = i & (DP - 1);
      float v = 0.f;
      if (d < DD)
        v = obs[(((size_t)(m * BATCH + b0 + r)) * TT + t) * DD + d];
      s_obs[i] = v;
    }
    __syncthreads();

    v8f acc[5];
    #pragma unroll
    for (int i = 0; i < 5; ++i) {
      const int nt = wave + i * 8;        // wave-uniform -> EXEC stays full
      if (nt < NTILES) {
        // ---- emission tile: E = obs_tile(16xDP) @ Bm^T(DPx16) ----
        const float* bB = Bpk + (((size_t)m * NTILES + nt) * KBB) * 64 + lane * 2;
        v8f e = {};
        #pragma unroll
        for (int kb = 0; kb < KBB; ++kb) {
          v2f a, b;
          a.x = s_obs[r16 * DP + kb * 4 + koff];
          a.y = s_obs[r16 * DP + kb * 4 + koff + 1];
          b = *(const v2f*)(bB + (size_t)kb * 64);
          e = __builtin_amdgcn_wmma_f32_16x16x4_f32(
              false, a, false, b, (short)0, e, false, false);
        }
        // ---- transition tile: R = alpha(16xQP) @ A(QPx16) (or pi at t=0) ----
        v8f rr;
        if (t == 0) {
          const float pv = piM[nt * 16 + r16];
          rr = (v8f){pv, pv, pv, pv, pv, pv, pv, pv};
        } else {
          const float* bA = Apk + (((size_t)m * NTILES + nt) * KBA) * 64 + lane * 2;
          v8f r8 = {};
          #pragma unroll 4
          for (int kb = 0; kb < KBA; ++kb) {
            if ((kb & 15) == 0)           // pull next panels toward the WGP
              __builtin_prefetch(bA + (size_t)(kb + 16) * 64, 0, 3);
            v2f a, b;
            a.x = s_alpha[r16 * QP + kb * 4 + koff];
            a.y = s_alpha[r16 * QP + kb * 4 + koff + 1];
            b = *(const v2f*)(bA + (size_t)kb * 64);
            r8 = __builtin_amdgcn_wmma_f32_16x16x4_f32(
                false, a, false, b, (short)0, r8, false, false);
          }
          rr = r8;
        }
        acc[i] = e * rr;                  // elementwise E_t * (alpha @ A)
      } else {
        acc[i] = (v8f){0.f, 0.f, 0.f, 0.f, 0.f, 0.f, 0.f, 0.f};
      }
    }
    __syncthreads();                      // all reads of s_alpha complete

    // scatter unnormalized new alpha back to LDS (C/D layout -> rows/cols)
    #pragma unroll
    for (int i = 0; i < 5; ++i) {
      const int nt = wave + i * 8;
      if (nt < NTILES) {
        const int col = nt * 16 + r16;
        #pragma unroll
        for (int v = 0; v < 8; ++v) {
          const int row = (lane < 16) ? v : (v + 8);
          s_alpha[row * QP + col] = acc[i][v];
        }
      }
    }
    __syncthreads();

    // per-row scale factor S = sum over states
    {
      const int r = tid >> 4, ii = tid & 15;
      float p = 0.f;
      for (int c = ii; c < QP; c += 16) p += s_alpha[r * QP + c];
      s_part[tid] = p;
    }
    __syncthreads();
    if (tid < 16) {
      float S = 0.f;
      #pragma unroll
      for (int j = 0; j < 16; ++j) S += s_part[tid * 16 + j];
      s_S[tid] = S;
      s_ll[tid] += __logf(S);             // loglik accumulates log(S_t)
    }
    __syncthreads();

    // normalize, emit log-unscaled forward variables, keep scaled alpha
    float* outT = out + (((size_t)t * MDL + m) * BATCH + b0) * QQ;
    for (int r = 0; r < 16; ++r) {
      const float invS = 1.f / s_S[r];
      const float llr  = s_ll[r];
      for (int c = tid; c < QQ; c += 256) {
        const float av = s_alpha[r * QP + c] * invS;
        s_alpha[r * QP + c] = av;
        outT[(size_t)r * QQ + c] = __logf(av + EPSF) + llr;
      }
    }
    __syncthreads();                      // alpha ready for next step
  }
}

// ---------------- launcher --------------------------------------------------
extern "C" void kernel_launch(void* const* d_in, const int* in_sizes, int n_in,
                              void* d_out, int out_size, void* d_ws, size_t ws_size,
                              hipStream_t stream) {
  const float* obs  = (const float*)d_in[0];   // [M,B,T,D]
  const float* Alog = (const float*)d_in[1];   // [M,q,q]
  const float* Blog = (const float*)d_in[2];   // [M,q,D]
  const float* Ilog = (const float*)d_in[3];   // [M,q]
  float* out = (float*)d_out;                  // [T,M,B,q]

  float* Apk = (float*)d_ws;                             // M*NTILES*KBA*64
  float* Bpk = Apk + (size_t)MDL * NTILES * KBA * 64;    // M*NTILES*KBB*64
  float* pip = Bpk + (size_t)MDL * NTILES * KBB * 64;    // M*QP
  (void)in_sizes; (void)n_in; (void)out_size; (void)ws_size;

  softmaxA_kernel <<<MDL * QP, 256, 0, stream>>>(Alog, Apk);
  softmaxB_kernel <<<MDL * QP, 32, 0, stream>>>(Blog, Bpk);
  softmaxPi_kernel<<<MDL, 256, 0, stream>>>(Ilog, pip);
  hmm_forward_kernel<<<MDL * (BATCH / 16), 256, 0, stream>>>(obs, Apk, Bpk, pip, out);
}